// GCNLayer_8443905704049
// MI455X (gfx1250) — compile-verified
//
#include <hip/hip_runtime.h>
#include <hip/hip_bf16.h>

// ---------------------------------------------------------------------------
// GCN layer for MI455X (gfx1250, wave32):
//   Phase 1: h = x @ W + b     -- fp32 WMMA (v_wmma_f32_16x16x4_f32)
//   Phase 2: out = A_sp @ h    -- per-edge wave gather + global_atomic_add_f32
//
// Roofline: GEMM is 13.1 GFLOP (negligible); SpMM gathers 3.28 GB from h
// (102.4 MB -> fully L2-resident in the 192 MB global L2) and scatters with
// f32 atomics into out (102.4 MB, also L2-resident). Whole problem is
// L2-bandwidth bound, so fp32 WMMA keeps reference precision at zero cost.
// ---------------------------------------------------------------------------

typedef __attribute__((ext_vector_type(2))) float v2f;
typedef __attribute__((ext_vector_type(8))) float v8f;

#define IN_F 256
#define OUT_F 256
#define NT_PER_WAVE 4   // N-tiles per wave: share one A load across 4 WMMAs

// One wave computes a 16x64 slab of h = x @ W + bias (4 adjacent 16x16 tiles,
// reusing the A operand). K consumed 4/WMMA -> 64 k-steps x 4 WMMAs.
__global__ __launch_bounds__(128) void gcn_gemm_wmma_f32(
    const float* __restrict__ x,      // [n_nodes, IN_F] row-major
    const float* __restrict__ w,      // [IN_F, OUT_F]  row-major
    const float* __restrict__ bias,   // [OUT_F]
    float* __restrict__ h,            // [n_nodes, OUT_F]
    int n_nodes)
{
    const int wave_in_blk = threadIdx.x >> 5;
    const int lane        = threadIdx.x & 31;
    const int slab_id     = blockIdx.x * 4 + wave_in_blk;

    const int m_tiles  = (n_nodes + 15) >> 4;
    const int n_groups = OUT_F / (16 * NT_PER_WAVE);            // = 4
    if (slab_id >= m_tiles * n_groups) return;                  // wave-uniform

    const int tile_m = slab_id >> 2;        // 16-row block of x/h
    const int n_base = (slab_id & 3) * 64;  // first of 4 16-col blocks

    const int half = lane >> 4;             // 0: lanes 0-15, 1: lanes 16-31
    const int lid  = lane & 15;

    // A (16x4 f32): lane lid holds row M=lid; half 0 -> K={0,1}, half 1 -> K={2,3}
    int row_a = tile_m * 16 + lid;
    if (row_a >= n_nodes) row_a = n_nodes - 1;   // clamp: EXEC must stay all-1s
    const float* arow = x + (size_t)row_a * IN_F + half * 2;

    // B (4x16 f32): lane lid holds col N=lid; half selects K pair {0,1}/{2,3}
    const float* bbase = w + (size_t)(half * 2) * OUT_F + n_base + lid;

    v8f acc[NT_PER_WAVE] = {};
#pragma unroll 4
    for (int k0 = 0; k0 < IN_F; k0 += 4) {
        v2f a;
        a.x = arow[k0];
        a.y = arow[k0 + 1];
#pragma unroll
        for (int nt = 0; nt < NT_PER_WAVE; ++nt) {
            v2f b;
            b.x = bbase[(size_t)k0 * OUT_F + nt * 16];
            b.y = bbase[(size_t)(k0 + 1) * OUT_F + nt * 16];
            // 8 args: (neg_a, A, neg_b, B, c_mod, C, reuse_a, reuse_b)
            acc[nt] = __builtin_amdgcn_wmma_f32_16x16x4_f32(
                false, a, false, b, (short)0, acc[nt], false, false);
        }
    }

    // C/D layout: VGPR i -> M = i + half*8, N = lid (per 16x16 tile)
    float* hrow = h + (size_t)(tile_m * 16 + half * 8) * OUT_F + n_base + lid;
    if (tile_m * 16 + 16 <= n_nodes) {
        // Full tile (always for n_nodes % 16 == 0): branch-free stores.
#pragma unroll
        for (int nt = 0; nt < NT_PER_WAVE; ++nt) {
            const float bv = bias[n_base + nt * 16 + lid];
#pragma unroll
            for (int i = 0; i < 8; ++i)
                hrow[(size_t)i * OUT_F + nt * 16] = acc[nt][i] + bv;
        }
    } else {
        // Ragged tail tile: per-row guard (divergent stores are legal).
#pragma unroll
        for (int nt = 0; nt < NT_PER_WAVE; ++nt) {
            const float bv = bias[n_base + nt * 16 + lid];
#pragma unroll
            for (int i = 0; i < 8; ++i) {
                const int row = tile_m * 16 + half * 8 + i;
                if (row < n_nodes)
                    hrow[(size_t)i * OUT_F + nt * 16] = acc[nt][i] + bv;
            }
        }
    }
}

// One wave per edge: gather h[col] (1 KB, L2-resident), scale by adj val,
// scatter-add into out[row] with f32 atomics (no-return -> STOREcnt only).
__global__ __launch_bounds__(256) void gcn_spmm_atomic(
    const float* __restrict__ h,
    const int*   __restrict__ edge_rows,
    const int*   __restrict__ edge_cols,
    const float* __restrict__ adj_vals,
    float* __restrict__ out,
    int n_edges)
{
    const int lane = threadIdx.x & 31;
    const int e    = blockIdx.x * 8 + (threadIdx.x >> 5);   // 8 waves / block
    if (e >= n_edges) return;                               // wave-uniform

    const int   r = edge_rows[e];   // wave-uniform -> scalar loads
    const int   c = edge_cols[e];
    const float a = adj_vals[e];

    const float4* hv = (const float4*)(h + (size_t)c * OUT_F);
    float*        op = out + (size_t)r * OUT_F;

    // 256 floats = 64 float4; lane covers float4 slots {lane, lane+32}
    const float4 v0 = hv[lane];
    const float4 v1 = hv[lane + 32];

    const int i0 = lane * 4;
    const int i1 = (lane + 32) * 4;
    atomicAdd(op + i0 + 0, a * v0.x);
    atomicAdd(op + i0 + 1, a * v0.y);
    atomicAdd(op + i0 + 2, a * v0.z);
    atomicAdd(op + i0 + 3, a * v0.w);
    atomicAdd(op + i1 + 0, a * v1.x);
    atomicAdd(op + i1 + 1, a * v1.y);
    atomicAdd(op + i1 + 2, a * v1.z);
    atomicAdd(op + i1 + 3, a * v1.w);
}

extern "C" void kernel_launch(void* const* d_in, const int* in_sizes, int n_in,
                              void* d_out, int out_size, void* d_ws, size_t ws_size,
                              hipStream_t stream) {
    const float* x    = (const float*)d_in[0];
    const int*   er   = (const int*)  d_in[1];
    const int*   ec   = (const int*)  d_in[2];
    const float* av   = (const float*)d_in[3];
    const float* w    = (const float*)d_in[4];
    const float* bias = (const float*)d_in[5];
    float*       out  = (float*)d_out;
    float*       h    = (float*)d_ws;     // [n_nodes, OUT_F] fp32 scratch

    const int n_nodes = in_sizes[0] / IN_F;
    const int n_edges = in_sizes[1];

    // out accumulates via atomics -> must start at zero (graph-safe memset).
    hipMemsetAsync(d_out, 0, (size_t)out_size * sizeof(float), stream);

    // Phase 1: dense transform, one wave per 16x64 slab, 4 waves per block.
    const int m_tiles     = (n_nodes + 15) / 16;
    const int total_slabs = m_tiles * (OUT_F / (16 * NT_PER_WAVE));
    const int gemm_blocks = (total_slabs + 3) / 4;
    gcn_gemm_wmma_f32<<<gemm_blocks, 128, 0, stream>>>(x, w, bias, h, n_nodes);

    // Phase 2: SpMM scatter, one wave per edge, 8 waves per block.
    const int spmm_blocks = (n_edges + 7) / 8;
    gcn_spmm_atomic<<<spmm_blocks, 256, 0, stream>>>(h, er, ec, av, out, n_edges);
}